// Attention_18279380812517
// MI455X (gfx1250) — compile-verified
//
#include <hip/hip_runtime.h>
#include <math.h>
#include <stdint.h>

#ifndef USE_ASYNC
#define USE_ASYNC 1   // gfx1250 global_load_async_to_lds_b128 staging
#endif

typedef float v2f __attribute__((ext_vector_type(2)));
typedef float v8f __attribute__((ext_vector_type(8)));

#define B_    4
#define N_    4096
#define C_    512
#define H_    8
#define HD_   64
#define FOLD_ 8
#define DP_   8
#define EPS_  1e-5f
#define NF    (N_ + 2)          // conv input rows incl. zero pad
#define KC    (3 * 2 * C_)      // 3072 = conv GEMM K
#define QCOLS (C_ + H_ * 16)    // 640 = q(512) + k-sel(64) + v-sel(64)

#define APITCH  36              // LDS pitch (floats) for A tiles, 16B-aligned
#define BPITCH2 136             // LDS pitch (floats) for K-pair-packed B tiles (2*64+8)
#define ATILE   (64 * APITCH)
#define BTILE   (16 * BPITCH2)

// ---------------- fold mean: u[b,n,f] = mean_j x[b,n,f*64+j] ----------------
__global__ void u_kernel(const float* __restrict__ x, float* __restrict__ u) {
  int idx = blockIdx.x * 256 + threadIdx.x;
  if (idx >= B_ * N_ * FOLD_) return;
  int f = idx & 7;
  long bn = idx >> 3;
  const float4* xr = (const float4*)(x + bn * C_ + f * (C_ / FOLD_));
  float s = 0.f;
  #pragma unroll
  for (int j = 0; j < (C_ / FOLD_) / 4; ++j) {
    float4 t = xr[j];
    s += t.x + t.y + t.z + t.w;
  }
  u[idx] = s * (1.0f / (C_ / FOLD_));
}

// ---------------- in-LDS radix-2 DIT FFT, length 4096 ----------------
__device__ void block_fft4096(float* re, float* im) {
  const int tid = threadIdx.x;  // 256 threads
  __syncthreads();
  for (int i = tid; i < N_; i += 256) {
    unsigned r = __brev((unsigned)i) >> (32 - 12);
    if (r > (unsigned)i) {
      float tr = re[i]; re[i] = re[r]; re[r] = tr;
      float ti = im[i]; im[i] = im[r]; im[r] = ti;
    }
  }
  __syncthreads();
  for (int s = 1; s <= 12; ++s) {
    const int half = 1 << (s - 1);
    const int mlen = 1 << s;
    const float ang = -6.28318530717958647692f / (float)mlen;
    for (int j = tid; j < (N_ >> 1); j += 256) {
      int grp = j >> (s - 1);
      int pos = j & (half - 1);
      int i0 = grp * mlen + pos;
      int i1 = i0 + half;
      float wr, wi;
      __sincosf(ang * (float)pos, &wi, &wr);
      float xr = re[i1], xi = im[i1];
      float tr = wr * xr - wi * xi;
      float ti = wr * xi + wi * xr;
      float ur = re[i0], ui = im[i0];
      re[i0] = ur + tr; im[i0] = ui + ti;
      re[i1] = ur - tr; im[i1] = ui - ti;
    }
    __syncthreads();
  }
}

// forward: U[b,f,:] = DFT_unnorm(u[b,:,f])
__global__ __launch_bounds__(256) void fft_fwd(const float* __restrict__ u,
                                               float2* __restrict__ U) {
  __shared__ float re[N_], im[N_];
  int b = blockIdx.x >> 3, f = blockIdx.x & 7;
  for (int i = threadIdx.x; i < N_; i += 256) {
    re[i] = u[((long)b * N_ + i) * FOLD_ + f];
    im[i] = 0.f;
  }
  block_fft4096(re, im);
  float2* Ub = U + ((long)(b * FOLD_ + f)) * N_;
  for (int i = threadIdx.x; i < N_; i += 256) { Ub[i].x = re[i]; Ub[i].y = im[i]; }
}

// inverse: m[b,:,c] = Re(IDFT(S))/N with S = U[b,c%8,:] * hermitian_ext(w[:,c])
// ifft via conj-trick: Re(IDFT_unnorm(S)) = Re(DFT_unnorm(conj(S)))
__global__ __launch_bounds__(256) void fft_inv(const float2* __restrict__ U,
                                               const float* __restrict__ wfft,
                                               float* __restrict__ mout) {
  __shared__ float re[N_], im[N_];
  int b = blockIdx.x >> 9, c = blockIdx.x & (C_ - 1);
  int f = c & 7;
  const float2* Ub = U + ((long)(b * FOLD_ + f)) * N_;
  for (int i = threadIdx.x; i < N_; i += 256) {
    float2 uv = Ub[i];
    float wr, wi;
    if (i <= N_ / 2) {
      wr = wfft[((long)i * C_ + c) * 2];
      wi = wfft[((long)i * C_ + c) * 2 + 1];
    } else {
      int k = N_ - i;
      wr =  wfft[((long)k * C_ + c) * 2];
      wi = -wfft[((long)k * C_ + c) * 2 + 1];
    }
    float sr = uv.x * wr - uv.y * wi;
    float si = uv.x * wi + uv.y * wr;
    re[i] = sr; im[i] = -si;   // conj(S)
  }
  block_fft4096(re, im);
  const float scale = 1.0f / (float)N_;  // 1/sqrt(N) (rfft ortho) * 1/sqrt(N) (irfft ortho)
  for (int i = threadIdx.x; i < N_; i += 256)
    mout[((long)b * N_ + i) * C_ + c] = re[i] * scale;
}

// ---------------- pack conv input (float4): inpP[b, n+1, :] = [m | x], pad rows 0 ----
__global__ void pack_inp(const float* __restrict__ m, const float* __restrict__ x,
                         float4* __restrict__ inpP) {
  long idx = (long)blockIdx.x * 256 + threadIdx.x;       // float4 index
  if (idx >= (long)B_ * NF * 256) return;
  int c4 = (int)(idx & 255);
  long bj = idx >> 8;
  int j = (int)(bj % NF);
  int b = (int)(bj / NF);
  float4 v = {0.f, 0.f, 0.f, 0.f};
  if (j > 0 && j < NF - 1) {
    long n = j - 1;
    v = (c4 < C_ / 4)
          ? ((const float4*)m)[((long)b * N_ + n) * (C_ / 4) + c4]
          : ((const float4*)x)[((long)b * N_ + n) * (C_ / 4) + (c4 - C_ / 4)];
  }
  inpP[idx] = v;
}

// Conv weights, K-pair interleaved: Wt[(k>>1)*2*C + co*2 + (k&1)] = conv_w[co][ci][kk]
// with k = kk*1024 + ci  (so conv-GEMM A rows are contiguous windows of inpP)
__global__ void pack_wconv(const float* __restrict__ conv_w, float* __restrict__ Wt) {
  int idx = blockIdx.x * 256 + threadIdx.x;
  if (idx >= KC * C_) return;
  int co = idx & (C_ - 1);
  int k  = idx >> 9;
  int kk = k >> 10;
  int ci = k & 1023;
  Wt[(long)(k >> 1) * (2 * C_) + co * 2 + (k & 1)] =
      conv_w[((long)co * 1024 + ci) * 3 + kk];
}

// QKV packed weights, K-pair interleaved: cols 0..511 = q rows;
// col 512+h*16+r = k-sel (r<8) / v-sel (r>=8)
__global__ void pack_wqkv(const float* __restrict__ qkv_w, const float* __restrict__ qkv_b,
                          const int* __restrict__ idx_right,
                          float* __restrict__ Wq2, float* __restrict__ qb2) {
  int idx = blockIdx.x * 256 + threadIdx.x;
  if (idx >= QCOLS * C_) return;
  int col = idx / C_;
  int k   = idx % C_;
  int row;
  if (col < C_) row = col;
  else {
    int j = col - C_;
    int h = j >> 4, jj = j & 15;
    row = (jj < 8) ? (C_ + h * HD_ + idx_right[jj])
                   : (2 * C_ + h * HD_ + idx_right[jj - 8]);
  }
  Wq2[(long)(k >> 1) * (2 * QCOLS) + col * 2 + (k & 1)] = qkv_w[(long)row * C_ + k];
  if (k == 0) qb2[col] = qkv_b[row];
}

// ---------------- GEMM tile staging: global -> LDS, 16B per lane ----------------
// B matrices are pre-packed K-pair interleaved, so 16B global chunks map 1:1 to LDS.
__device__ __forceinline__ void gemm_stage(
    const float* __restrict__ A, const float* __restrict__ Bmat,
    float* __restrict__ As, float* __restrict__ Bs,
    int row0, int col0, int k0, int lda, int Mb, int bstride, int ncols, int tid) {
  #pragma unroll
  for (int i = 0; i < 2; ++i) {            // A tile: 64 rows x 32 k = 512 float4
    int e = tid + i * 256;
    int r = e >> 3, kq = e & 7;
    int rowG = row0 + r;
    const float* gp = A + (long)rowG * lda + (long)(rowG / Mb) * bstride + k0 + kq * 4;
#if USE_ASYNC
    uint32_t lo = (uint32_t)(uintptr_t)(&As[r * APITCH + kq * 4]);
    asm volatile("global_load_async_to_lds_b128 %0, %1, off"
                 :: "v"(lo), "v"(gp) : "memory");
#else
    *(float4*)&As[r * APITCH + kq * 4] = *(const float4*)gp;
#endif
  }
  #pragma unroll
  for (int i = 0; i < 2; ++i) {            // B tile: 16 kp x 64 cols x 2 = 512 float4
    int e = tid + i * 256;
    int kp = e >> 5, cp = e & 31;          // cp = column pair
    const float* gp = Bmat + (long)(k0 / 2 + kp) * (2 * ncols) + (col0 + cp * 2) * 2;
#if USE_ASYNC
    uint32_t lo = (uint32_t)(uintptr_t)(&Bs[kp * BPITCH2 + cp * 4]);
    asm volatile("global_load_async_to_lds_b128 %0, %1, off"
                 :: "v"(lo), "v"(gp) : "memory");
#else
    *(float4*)&Bs[kp * BPITCH2 + cp * 4] = *(const float4*)gp;
#endif
  }
}

// ---------------- WMMA f32 16x16x4 GEMM: C = A(MxK) * B(KxN) + bias ----------------
// A rows: ptr = A + rowG*lda + (rowG/Mb)*bstride  (handles per-batch pad rows; rows overlap OK)
// Double-buffered LDS tiles; async DMA of tile t+1 overlaps WMMA of tile t.
// Both A and B fragments are single ds_load_b64 (B is K-pair packed).
__global__ __launch_bounds__(256) void gemm_f32_wmma(
    const float* __restrict__ A, const float* __restrict__ Bmat,
    const float* __restrict__ bias, float* __restrict__ Cmat,
    int K, int lda, int Mb, int bstride, int ncols, int ldo) {
  __shared__ float Asb[2 * ATILE];
  __shared__ float Bsb[2 * BTILE];
  const int row0 = blockIdx.y * 64;
  const int col0 = blockIdx.x * 64;
  const int tid  = threadIdx.x;
  const int lane = tid & 31;
  const int wv   = tid >> 5;
  const int half = lane >> 4;    // K-half select per ISA A/B f32 layout
  const int l16  = lane & 15;

  v8f acc0 = {0, 0, 0, 0, 0, 0, 0, 0};
  v8f acc1 = {0, 0, 0, 0, 0, 0, 0, 0};
  const int t0  = wv * 2;                 // 2 of the 16 16x16 tiles per wave
  const int mi  = t0 >> 2;
  const int ni0 = t0 & 3;
  const int ni1 = ni0 + 1;

  gemm_stage(A, Bmat, Asb, Bsb, row0, col0, 0, lda, Mb, bstride, ncols, tid);

  for (int k0 = 0; k0 < K; k0 += 32) {
    const int cur = (k0 >> 5) & 1;
    const float* Ac = Asb + cur * ATILE;
    const float* Bc = Bsb + cur * BTILE;
#if USE_ASYNC
    asm volatile("s_wait_asynccnt 0x0" ::: "memory");
#endif
    __syncthreads();                      // tile `cur` visible to all waves
    if (k0 + 32 < K)                      // prefetch next tile into other buffer
      gemm_stage(A, Bmat, Asb + (cur ^ 1) * ATILE, Bsb + (cur ^ 1) * BTILE,
                 row0, col0, k0 + 32, lda, Mb, bstride, ncols, tid);
    #pragma unroll
    for (int kk4 = 0; kk4 < 8; ++kk4) {
      int kp = kk4 * 2 + half;            // lanes 0-15: K {0,1}; 16-31: K {2,3}
      v2f a0 = *(const v2f*)&Ac[(mi * 16 + l16) * APITCH + kp * 2];
      v2f b0 = *(const v2f*)&Bc[kp * BPITCH2 + (ni0 * 16 + l16) * 2];
      acc0 = __builtin_amdgcn_wmma_f32_16x16x4_f32(false, a0, false, b0,
                                                   (short)0, acc0, false, false);
      v2f b1 = *(const v2f*)&Bc[kp * BPITCH2 + (ni1 * 16 + l16) * 2];
      acc1 = __builtin_amdgcn_wmma_f32_16x16x4_f32(false, a0, false, b1,
                                                   (short)0, acc1, false, false);
    }
  }
  #pragma unroll
  for (int i = 0; i < 8; ++i) {
    int r = mi * 16 + i + 8 * half;       // C/D layout: VGPR i -> rows {i, i+8}
    int c0 = ni0 * 16 + l16;
    int c1 = ni1 * 16 + l16;
    Cmat[(long)(row0 + r) * ldo + col0 + c0] = acc0[i] + bias[col0 + c0];
    Cmat[(long)(row0 + r) * ldo + col0 + c1] = acc1[i] + bias[col0 + c1];
  }
}

// ---------------- BN over (batch, channel) per position n ----------------
__global__ void bn_stats(const float* __restrict__ h, float* __restrict__ mu,
                         float* __restrict__ rv) {
  __shared__ float s1[256], s2[256];
  int n = blockIdx.x;
  float a = 0.f, b = 0.f;
  for (int i = threadIdx.x; i < B_ * C_; i += 256) {
    int bb = i >> 9, c = i & (C_ - 1);
    float v = h[((long)bb * N_ + n) * C_ + c];
    a += v; b += v * v;
  }
  s1[threadIdx.x] = a; s2[threadIdx.x] = b;
  __syncthreads();
  for (int s = 128; s > 0; s >>= 1) {
    if (threadIdx.x < s) { s1[threadIdx.x] += s1[threadIdx.x + s]; s2[threadIdx.x] += s2[threadIdx.x + s]; }
    __syncthreads();
  }
  if (threadIdx.x == 0) {
    float m = s1[0] * (1.0f / (B_ * C_));
    float var = s2[0] * (1.0f / (B_ * C_)) - m * m;
    mu[n] = m;
    rv[n] = rsqrtf(var + EPS_);
  }
}

__global__ void bn_elu(const float4* __restrict__ h, const float* __restrict__ mu,
                       const float* __restrict__ rv, const float* __restrict__ g,
                       const float* __restrict__ bta, float4* __restrict__ xh) {
  long idx = (long)blockIdx.x * 256 + threadIdx.x;       // float4 index
  if (idx >= (long)B_ * N_ * (C_ / 4)) return;
  int n = (int)((idx >> 7) % N_);                        // C_/4 = 128 vec4 per row
  float sc = rv[n] * g[n];
  float mb = bta[n] - mu[n] * sc;
  float4 hv = h[idx];
  float4 o;
  o.x = hv.x * sc + mb; o.x = o.x > 0.f ? o.x : expm1f(o.x);
  o.y = hv.y * sc + mb; o.y = o.y > 0.f ? o.y : expm1f(o.y);
  o.z = hv.z * sc + mb; o.z = o.z > 0.f ? o.z : expm1f(o.z);
  o.w = hv.w * sc + mb; o.w = o.w > 0.f ? o.w : expm1f(o.w);
  xh[idx] = o;
}

// ---------------- left K/V at DP gathered positions (tiny GEMM) ----------------
__global__ void kv_left(const float* __restrict__ xh, const float* __restrict__ qkv_w,
                        const float* __restrict__ qkv_b, const int* __restrict__ idx_left,
                        float* __restrict__ k1, float* __restrict__ v1) {
  __shared__ float xr[C_];
  int b = blockIdx.x >> 3, r = blockIdx.x & 7;
  int n = idx_left[r];
  for (int i = threadIdx.x; i < C_; i += 256)
    xr[i] = xh[((long)b * N_ + n) * C_ + i];
  __syncthreads();
  for (int o = threadIdx.x; o < 2 * C_; o += 256) {
    int wrow = C_ + o;   // k rows: 512+o (o<512); v rows: 1024+(o-512) == 512+o
    float s = qkv_b[wrow];
    const float4* wp = (const float4*)(qkv_w + (long)wrow * C_);
    const float4* xp = (const float4*)xr;
    for (int k = 0; k < C_ / 4; ++k) {
      float4 wv = wp[k], xv = xp[k];
      s += xv.x * wv.x + xv.y * wv.y + xv.z * wv.z + xv.w * wv.w;
    }
    if (o < C_) {
      int h = o >> 6, d = o & 63;
      k1[(((long)b * H_ + h) * DP_ + r) * HD_ + d] = s;
    } else {
      int oo = o - C_;
      int h = oo >> 6, d = oo & 63;
      v1[(((long)b * H_ + h) * DP_ + r) * HD_ + d] = s;
    }
  }
}

// ---------------- right attention: reduce over n, softmax over r, swap ----------------
__global__ __launch_bounds__(256) void right_dots(const float* __restrict__ qbuf,
                                                  float* __restrict__ attnr) {
  __shared__ float qch[64 * 68];
  __shared__ float kch[64 * 8];
  __shared__ float dsh[512];
  int b = blockIdx.x >> 3, h = blockIdx.x & 7;
  int tid = threadIdx.x;
  int d0 = tid >> 3, r0 = tid & 7;
  int p1 = tid + 256, d1 = p1 >> 3, r1 = p1 & 7;
  float acc0 = 0.f, acc1 = 0.f;
  for (int nc = 0; nc < N_; nc += 64) {
    __syncthreads();
    for (int i = tid; i < 64 * 16; i += 256) {       // q chunk as float4
      int nn = i >> 4, dq = i & 15;
      *(float4*)&qch[nn * 68 + dq * 4] =
          *(const float4*)&qbuf[((long)b * N_ + nc + nn) * QCOLS + h * HD_ + dq * 4];
    }
    for (int i = tid; i < 64 * 2; i += 256) {        // k-sel chunk as float4
      int nn = i >> 1, rq = i & 1;
      *(float4*)&kch[nn * 8 + rq * 4] =
          *(const float4*)&qbuf[((long)b * N_ + nc + nn) * QCOLS + C_ + h * 16 + rq * 4];
    }
    __syncthreads();
    #pragma unroll 4
    for (int nn = 0; nn < 64; ++nn) {
      acc0 += qch[nn * 68 + d0] * kch[nn * 8 + r0];
      acc1 += qch[nn * 68 + d1] * kch[nn * 8 + r1];
    }
  }
  dsh[tid]       = acc0 * (1.0f / 64.0f);   // 1/sqrt(N)
  dsh[tid + 256] = acc1 * (1.0f / 64.0f);
  __syncthreads();
  if (tid < 64) {
    int d = tid;
    float mx = -1e30f;
    #pragma unroll
    for (int r = 0; r < 8; ++r) mx = fmaxf(mx, dsh[d * 8 + r]);
    float e[8], sum = 0.f;
    #pragma unroll
    for (int r = 0; r < 8; ++r) { e[r] = __expf(dsh[d * 8 + r] - mx); sum += e[r]; }
    float inv = 1.f / sum;
    #pragma unroll
    for (int r = 0; r < 8; ++r)
      attnr[(((long)b * H_ + h) * DP_ + r) * HD_ + d] = e[r] * inv;
  }
}

// ---------------- fused: left attention + right apply + dual LayerNorm ----------------
__global__ __launch_bounds__(512) void final_kernel(
    const float* __restrict__ qbuf, const float* __restrict__ k1,
    const float* __restrict__ v1, const float* __restrict__ attnr,
    const float* __restrict__ ln1g, const float* __restrict__ ln1b,
    const float* __restrict__ ln2g, const float* __restrict__ ln2b,
    float* __restrict__ out) {
  __shared__ float qrow[QCOLS];
  __shared__ float k1s[H_ * DP_ * HD_];
  __shared__ float v1s[H_ * DP_ * HD_];
  __shared__ float ars[H_ * DP_ * HD_];
  __shared__ float red1[512], red2[512];
  int b = blockIdx.x >> 12;
  int n = blockIdx.x & (N_ - 1);
  int t = threadIdx.x;
  for (int i = t; i < QCOLS; i += 512)
    qrow[i] = qbuf[((long)b * N_ + n) * QCOLS + i];
  for (int i = t; i < H_ * DP_ * HD_; i += 512) {
    k1s[i] = k1[(long)b * H_ * DP_ * HD_ + i];
    v1s[i] = v1[(long)b * H_ * DP_ * HD_ + i];
    ars[i] = attnr[(long)b * H_ * DP_ * HD_ + i];
  }
  __syncthreads();
  int h = t >> 6, d = t & 63;
  // left: softmax_r( q.k1 / sqrt(HD) ) @ v1
  float logit[8];
  float mx = -1e30f;
  const float* qh = &qrow[h * HD_];
  #pragma unroll
  for (int r = 0; r < 8; ++r) {
    const float* kr = &k1s[(h * DP_ + r) * HD_];
    float s = 0.f;
    #pragma unroll 8
    for (int dd = 0; dd < HD_; ++dd) s += qh[dd] * kr[dd];
    logit[r] = s * 0.125f;                   // 1/sqrt(HD)
    mx = fmaxf(mx, logit[r]);
  }
  float sum = 0.f;
  #pragma unroll
  for (int r = 0; r < 8; ++r) { logit[r] = __expf(logit[r] - mx); sum += logit[r]; }
  float inv = 1.f / sum;
  float lv = 0.f;
  #pragma unroll
  for (int r = 0; r < 8; ++r) lv += logit[r] * inv * v1s[(h * DP_ + r) * HD_ + d];
  // right: sum_r v2[b,n,h,r] * attn_r[b,h,r,d]
  float rvv = 0.f;
  #pragma unroll
  for (int r = 0; r < 8; ++r)
    rvv += qrow[C_ + h * 16 + 8 + r] * ars[(h * DP_ + r) * HD_ + d];
  // LayerNorm(left)
  red1[t] = lv; red2[t] = lv * lv;
  __syncthreads();
  for (int s = 256; s > 0; s >>= 1) {
    if (t < s) { red1[t] += red1[t + s]; red2[t] += red2[t + s]; }
    __syncthreads();
  }
  float mu1 = red1[0] * (1.f / 512.f);
  float var1 = red2[0] * (1.f / 512.f) - mu1 * mu1;
  float o1 = (lv - mu1) * rsqrtf(var1 + EPS_) * ln1g[t] + ln1b[t];
  __syncthreads();
  // LayerNorm(right)
  red1[t] = rvv; red2[t] = rvv * rvv;
  __syncthreads();
  for (int s = 256; s > 0; s >>= 1) {
    if (t < s) { red1[t] += red1[t + s]; red2[t] += red2[t + s]; }
    __syncthreads();
  }
  float mu2 = red1[0] * (1.f / 512.f);
  float var2 = red2[0] * (1.f / 512.f) - mu2 * mu2;
  float o2 = (rvv - mu2) * rsqrtf(var2 + EPS_) * ln2g[t] + ln2b[t];
  out[((long)b * N_ + n) * C_ + t] = 0.5f * (o1 + o2);
}

extern "C" void kernel_launch(void* const* d_in, const int* in_sizes, int n_in,
                              void* d_out, int out_size, void* d_ws, size_t ws_size,
                              hipStream_t stream) {
  const float* x       = (const float*)d_in[0];
  const float* wfft    = (const float*)d_in[1];
  const float* conv_w  = (const float*)d_in[2];
  const float* conv_b  = (const float*)d_in[3];
  const float* bn_g    = (const float*)d_in[4];
  const float* bn_b    = (const float*)d_in[5];
  const float* qkv_w   = (const float*)d_in[6];
  const float* qkv_b   = (const float*)d_in[7];
  const float* ln1g    = (const float*)d_in[8];
  const float* ln1b    = (const float*)d_in[9];
  const float* ln2g    = (const float*)d_in[10];
  const float* ln2b    = (const float*)d_in[11];
  const int* idx_left  = (const int*)d_in[12];
  const int* idx_right = (const int*)d_in[13];
  float* out = (float*)d_out;
  (void)in_sizes; (void)n_in; (void)out_size; (void)ws_size;

  char* wsb = (char*)d_ws;
  size_t off = 0;
  auto alloc = [&](size_t nfloats) {
    float* p = (float*)(wsb + off);
    off += ((nfloats * 4 + 255) / 256) * 256;
    return p;
  };
  float*  u     = alloc((size_t)B_ * N_ * FOLD_);
  float2* U     = (float2*)alloc((size_t)B_ * FOLD_ * N_ * 2);
  float*  mxh   = alloc((size_t)B_ * N_ * C_);      // m, later reused as xh
  float*  inpq  = alloc((size_t)B_ * NF * 1024);    // conv input, later reused as qbuf
  float*  Wt    = alloc((size_t)KC * C_);
  float*  hbuf  = alloc((size_t)B_ * N_ * C_);
  float*  mu    = alloc(N_);
  float*  rv    = alloc(N_);
  float*  Wq2   = alloc((size_t)C_ * QCOLS);
  float*  qb2   = alloc(QCOLS);
  float*  k1    = alloc((size_t)B_ * H_ * DP_ * HD_);
  float*  v1    = alloc((size_t)B_ * H_ * DP_ * HD_);
  float*  attnr = alloc((size_t)B_ * H_ * DP_ * HD_);

  u_kernel<<<(B_ * N_ * FOLD_ + 255) / 256, 256, 0, stream>>>(x, u);
  fft_fwd<<<B_ * FOLD_, 256, 0, stream>>>(u, U);
  fft_inv<<<B_ * C_, 256, 0, stream>>>(U, wfft, mxh);

  long pin = (long)B_ * NF * 256;    // float4 count
  pack_inp<<<(int)((pin + 255) / 256), 256, 0, stream>>>(mxh, x, (float4*)inpq);
  pack_wconv<<<(KC * C_ + 255) / 256, 256, 0, stream>>>(conv_w, Wt);

  // conv as GEMM: M=16384, K=3072, N=512; per-batch extra row stride 2*1024 (pad rows)
  gemm_f32_wmma<<<dim3(C_ / 64, (B_ * N_) / 64), 256, 0, stream>>>(
      inpq, Wt, conv_b, hbuf, KC, 1024, N_, 2048, C_, C_);

  bn_stats<<<N_, 256, 0, stream>>>(hbuf, mu, rv);
  bn_elu<<<((B_ * N_ * (C_ / 4)) + 255) / 256, 256, 0, stream>>>(
      (const float4*)hbuf, mu, rv, bn_g, bn_b, (float4*)mxh);

  pack_wqkv<<<(QCOLS * C_ + 255) / 256, 256, 0, stream>>>(qkv_w, qkv_b, idx_right, Wq2, qb2);
  // q + gathered k/v dims: M=16384, K=512, N=640
  gemm_f32_wmma<<<dim3(QCOLS / 64, (B_ * N_) / 64), 256, 0, stream>>>(
      mxh, Wq2, qb2, inpq, C_, C_, N_, 0, QCOLS, QCOLS);

  kv_left<<<B_ * DP_, 256, 0, stream>>>(mxh, qkv_w, qkv_b, idx_left, k1, v1);
  right_dots<<<B_ * H_, 256, 0, stream>>>(inpq, attnr);
  final_kernel<<<B_ * N_, 512, 0, stream>>>(inpq, k1, v1, attnr,
                                            ln1g, ln1b, ln2g, ln2b, out);
}